// BinsChamferLoss_43894565765368
// MI455X (gfx1250) — compile-verified
//
#include <hip/hip_runtime.h>

typedef __attribute__((ext_vector_type(2))) float v2f;
typedef __attribute__((ext_vector_type(8))) float v8f;

#define N_SAMPLES 8
#define P_BINS    256
#define M_PTS     65536
#define BIGF      1e10f
#define MINV      1e-8f
#define WAVES     8     // waves per block (256 threads)
#define TT        8     // 16-target tiles per wave
#define BT        16    // 16-bin tiles (256 bins)

// ---------------------------------------------------------------------------
// Workspace init (harness does not re-zero between replays)
// ---------------------------------------------------------------------------
__global__ void chamfer_init(unsigned* __restrict__ binMin,
                             float* __restrict__ sumY,
                             unsigned* __restrict__ cntY) {
    int idx = blockIdx.x * 256 + threadIdx.x;
    if (idx < N_SAMPLES * P_BINS) binMin[idx] = __float_as_uint(BIGF);
    if (idx < N_SAMPLES) { sumY[idx] = 0.f; cntY[idx] = 0u; }
}

// ---------------------------------------------------------------------------
// Main: pairwise squared distances via V_WMMA_F32_16X16X4_F32.
//   D[m,j] = bc_m*(-2 t_j) + bc_m^2*1 + 1*t_j^2 = (bc_m - t_j)^2
// A[m,:] = (bc, bc^2, 1, 0)   B[:,j] = (-2t, 1, t^2, 0)^T   C = 0
// 16x4 f32 A layout: lanes 0-15 hold K={0,1}, lanes 16-31 hold K={2,3};
// B mirrors it. All fragments built branchlessly (v_cndmask) and preloaded
// so the 128-WMMA hot loop has no LDS traffic or EXEC churn.
// ---------------------------------------------------------------------------
__global__ __launch_bounds__(256)
void chamfer_main(const float* __restrict__ bins,
                  const float* __restrict__ tmap,
                  unsigned* __restrict__ binMin,
                  float* __restrict__ sumY,
                  unsigned* __restrict__ cntY) {
    __shared__ float    s_bc[P_BINS];
    __shared__ unsigned s_min[P_BINS];

    const int  tid  = threadIdx.x;
    const int  lane = tid & 31;
    const int  wave = tid >> 5;
    const int  n    = blockIdx.y;
    const int  col  = lane & 15;
    const bool hi   = lane >= 16;

    // bin centers -> LDS; per-bin block-local min accumulator
    {
        float b0 = bins[n * (P_BINS + 1) + tid];
        float b1 = bins[n * (P_BINS + 1) + tid + 1];
        s_bc[tid]  = 0.5f * (b0 + b1);
        s_min[tid] = __float_as_uint(BIGF);
    }
    __syncthreads();

    // Preload all 16 A fragments (branchless: cndmask, no divergent ds_load)
    v2f Af[BT];
#pragma unroll
    for (int bt = 0; bt < BT; ++bt) {
        float bc = s_bc[bt * 16 + col];         // unconditional LDS read
        v2f a;
        a.x = hi ? 1.f : bc;                    // cols K=0/K=2
        a.y = hi ? 0.f : bc * bc;               // cols K=1/K=3
        Af[bt] = a;
    }

    const float* t    = tmap + (size_t)n * M_PTS;
    const int    base = (blockIdx.x * WAVES + wave) * (TT * 16);

    // Preload B fragments for this wave's 8 target tiles
    v2f      Bf[TT];
    float    accY[TT];
    unsigned maskbits = 0;
#pragma unroll
    for (int k = 0; k < TT; ++k) {
        float tv = t[base + k * 16 + col];
        if (tv >= MINV) maskbits |= (1u << k);
        v2f b;
        b.x = hi ? tv * tv : -2.f * tv;         // rows K=0/K=2
        b.y = hi ? 0.f     : 1.f;               // rows K=1/K=3
        Bf[k]   = b;
        accY[k] = BIGF;
    }

    // Sweep all 16 bin tiles (256 bins) against the wave's 128 targets
    for (int bt = 0; bt < BT; ++bt) {
        float accX[8];
#pragma unroll
        for (int r = 0; r < 8; ++r) accX[r] = BIGF;

#pragma unroll
        for (int k = 0; k < TT; ++k) {
            v8f c = {};
            v8f D = __builtin_amdgcn_wmma_f32_16x16x4_f32(
                        false, Af[bt], false, Bf[k], (short)0, c, false, false);

            // cham_y path: min over this tile's 8 bin rows (unmasked d)
            float mn = fminf(fminf(fminf(D[0], D[1]), fminf(D[2], D[3])),
                             fminf(fminf(D[4], D[5]), fminf(D[6], D[7])));
            accY[k] = fminf(accY[k], mn);

            // cham_x path: mask invalid targets with BIG, min over targets
            bool m = (maskbits >> k) & 1u;
#pragma unroll
            for (int r = 0; r < 8; ++r)
                accX[r] = fminf(accX[r], m ? D[r] : BIGF);
        }

        // Reduce accX across the 16 target-columns of each half-wave
#pragma unroll
        for (int r = 0; r < 8; ++r) {
            float v = accX[r];
            v = fminf(v, __shfl_xor(v, 1, 32));
            v = fminf(v, __shfl_xor(v, 2, 32));
            v = fminf(v, __shfl_xor(v, 4, 32));
            v = fminf(v, __shfl_xor(v, 8, 32));
            if (col == 0)   // lane0 -> rows 0..7, lane16 -> rows 8..15
                atomicMin(&s_min[bt * 16 + r + (hi ? 8 : 0)], __float_as_uint(v));
        }
    }

    // cham_y: fold half-waves, masked sum + count, one atomic per wave
    float    sY = 0.f;
    unsigned cY = 0;
#pragma unroll
    for (int k = 0; k < TT; ++k) {
        float v = fminf(accY[k], __shfl_xor(accY[k], 16, 32));
        bool  m = (maskbits >> k) & 1u;
        sY += (!hi && m) ? v : 0.f;
        cY += (!hi && m) ? 1u : 0u;
    }
#pragma unroll
    for (int s = 1; s < 32; s <<= 1) {
        sY += __shfl_xor(sY, s, 32);
        cY += __shfl_xor(cY, s, 32);
    }
    if (lane == 0) {
        atomicAdd(&sumY[n], sY);
        atomicAdd(&cntY[n], cY);
    }

    // Flush block-local per-bin mins to global (bit-pattern atomicMin, d >= 0)
    __syncthreads();
    atomicMin(&binMin[n * P_BINS + tid], s_min[tid]);
}

// ---------------------------------------------------------------------------
// Finalize: wave n reduces its 256 bin-mins; scalar mean over samples
// ---------------------------------------------------------------------------
__global__ __launch_bounds__(256)
void chamfer_final(const unsigned* __restrict__ binMin,
                   const float* __restrict__ sumY,
                   const unsigned* __restrict__ cntY,
                   float* __restrict__ out) {
    __shared__ float s_res[N_SAMPLES];
    const int lane = threadIdx.x & 31;
    const int n    = threadIdx.x >> 5;

    float s = 0.f;
#pragma unroll
    for (int k = 0; k < P_BINS / 32; ++k)
        s += __uint_as_float(binMin[n * P_BINS + k * 32 + lane]);
#pragma unroll
    for (int sh = 1; sh < 32; sh <<= 1) s += __shfl_xor(s, sh, 32);

    if (lane == 0) {
        float chamx = s / (float)P_BINS;
        float chamy = sumY[n] / (float)cntY[n];
        s_res[n] = chamx + chamy;
    }
    __syncthreads();
    if (threadIdx.x == 0) {
        float acc = 0.f;
        for (int i = 0; i < N_SAMPLES; ++i) acc += s_res[i];
        out[0] = acc / (float)N_SAMPLES;
    }
}

// ---------------------------------------------------------------------------
extern "C" void kernel_launch(void* const* d_in, const int* in_sizes, int n_in,
                              void* d_out, int out_size, void* d_ws, size_t ws_size,
                              hipStream_t stream) {
    (void)in_sizes; (void)n_in; (void)out_size; (void)ws_size;

    const float* bins = (const float*)d_in[0];   // [8, 257]
    const float* tmap = (const float*)d_in[1];   // [8, 256, 256]

    unsigned* binMin = (unsigned*)d_ws;                               // 8*256 u32
    float*    sumY   = (float*)((char*)d_ws + N_SAMPLES * P_BINS * 4);
    unsigned* cntY   = (unsigned*)((char*)sumY + N_SAMPLES * 4);

    chamfer_init<<<dim3((N_SAMPLES * P_BINS + 255) / 256), 256, 0, stream>>>(
        binMin, sumY, cntY);

    dim3 grid(M_PTS / (WAVES * TT * 16), N_SAMPLES);   // (64, 8)
    chamfer_main<<<grid, 256, 0, stream>>>(bins, tmap, binMin, sumY, cntY);

    chamfer_final<<<1, 256, 0, stream>>>(binMin, sumY, cntY, (float*)d_out);
}